// OptimizedQuantumLayer_38577396253278
// MI455X (gfx1250) — compile-verified
//
#include <hip/hip_runtime.h>
#include <hip/hip_bf16.h>
#include <math.h>

typedef __attribute__((ext_vector_type(2))) float v2f;
typedef __attribute__((ext_vector_type(8))) float v8f;

#define N_QUBITS 10
#define N_LAYERS 3
#define WAVES_PER_BLOCK 4
#define STATE_SZ 1024

// RY(t) = [[c, -s],[s, c]], c = cos(t/2), s = sin(t/2)
__device__ __forceinline__ float ry_elem(float c, float s, int row, int col) {
  return (row == col) ? c : ((row == 0) ? -s : s);
}

// Map (column, group-value) -> flat state index.
// MODE 0: group occupies bits 9..6 (wires 0-3); columns are bits 5..0.
// MODE 1: group occupies bits 5..2 (wires 4-7); columns are bits {9..6, 1..0}.
template <int MODE>
__device__ __forceinline__ int map_idx(int col, int g) {
  if (MODE == 0) return (g << 6) | col;
  return ((col >> 2) << 6) | (g << 2) | (col & 3);
}

// Apply a 4-qubit grouped RY gate G (16x16, tensor product of four 2x2 RYs)
// to the state in LDS: state' = G x state_reshaped(16 x 64).
// Executed as 4 column tiles x 4 K-chunks of V_WMMA_F32_16X16X4_F32.
template <int MODE>
__device__ __forceinline__ void apply_group4(float* __restrict__ S,
                                             const float* c4, const float* s4,
                                             int lane) {
  const int mrow  = lane & 15;        // A: M row / B,D: N column
  const int khalf = (lane >> 4) << 1; // lane half selects K pair {0,1} vs {2,3}
#pragma unroll
  for (int t = 0; t < 4; ++t) {
    v8f acc = {0.f, 0.f, 0.f, 0.f, 0.f, 0.f, 0.f, 0.f};
#pragma unroll
    for (int kk = 0; kk < 4; ++kk) {
      v2f a, b;
#pragma unroll
      for (int v = 0; v < 2; ++v) {
        const int kcol = kk * 4 + khalf + v;
        // G[m][k] = prod_j M_{w_j}[bit_j(m)][bit_j(k)]  (j=0 is MSB of group)
        float e = 1.0f;
#pragma unroll
        for (int j = 0; j < 4; ++j)
          e *= ry_elem(c4[j], s4[j], (mrow >> (3 - j)) & 1, (kcol >> (3 - j)) & 1);
        a[v] = e;
        // B[k][n] = S[index(col = t*16 + n, group = k)]
        b[v] = S[map_idx<MODE>(t * 16 + mrow, kcol)];
      }
      acc = __builtin_amdgcn_wmma_f32_16x16x4_f32(false, a, false, b,
                                                  (short)0, acc, false, false);
    }
    // D layout: VGPR r holds rows (r + 8*laneHalf), column = lane&15
    const int rbase = (lane >> 4) << 3;
#pragma unroll
    for (int r = 0; r < 8; ++r)
      S[map_idx<MODE>(t * 16 + mrow, rbase + r)] = acc[r];
  }
}

// Apply the 2-qubit grouped RY (wires 8,9 -> state bits 1..0): state' has
// out[(col<<2)|m] = sum_k G2[m][k] * S[(col<<2)|k], cols = 256, K = 4 exactly.
// A rows replicated (m & 3) so every WMMA column is useful; keep rows 0..3.
__device__ __forceinline__ void apply_group2(float* __restrict__ S,
                                             float c8, float s8,
                                             float c9, float s9, int lane) {
  const int mrow  = lane & 15;
  const int khalf = (lane >> 4) << 1;
  v2f a;
#pragma unroll
  for (int v = 0; v < 2; ++v) {
    const int k = khalf + v;
    const int g = mrow & 3; // bit1 <-> wire 8, bit0 <-> wire 9
    a[v] = ry_elem(c8, s8, (g >> 1) & 1, (k >> 1) & 1) *
           ry_elem(c9, s9, g & 1, k & 1);
  }
#pragma unroll
  for (int t = 0; t < 16; ++t) {
    const int ncol = t * 16 + mrow;
    v2f b;
#pragma unroll
    for (int v = 0; v < 2; ++v)
      b[v] = S[(ncol << 2) | (khalf + v)];
    v8f acc = {0.f, 0.f, 0.f, 0.f, 0.f, 0.f, 0.f, 0.f};
    acc = __builtin_amdgcn_wmma_f32_16x16x4_f32(false, a, false, b,
                                                (short)0, acc, false, false);
    if (lane < 16) { // rows 0..3 live in VGPRs 0..3 of lanes 0..15
#pragma unroll
      for (int r = 0; r < 4; ++r)
        S[(ncol << 2) | r] = acc[r];
    }
  }
}

// Composed source index for one CNOT ring layer:
// gates applied in order CNOT(0,1)..CNOT(8,9), CNOT(9,0);
// new[i] = old[f1(f2(...f10(i)))], f = i ^ (ctrlbit(i) ? tgt_mask : 0).
__device__ __forceinline__ int cnot_src(int i) {
  int s = i;
  s ^= (s & 1) << 9; // CNOT(9,0): ctrl bit0, tgt bit9 (applied last -> first here)
#pragma unroll
  for (int w = 8; w >= 0; --w)
    s ^= ((s >> (9 - w)) & 1) << (8 - w); // CNOT(w,w+1): ctrl bit 9-w, tgt 8-w
  return s;
}

__global__ __launch_bounds__(WAVES_PER_BLOCK * 32, 1)
void qlayer_wmma_kernel(const float* __restrict__ x,
                        const float* __restrict__ params,
                        float* __restrict__ out, int batch) {
  __shared__ float lds[WAVES_PER_BLOCK * STATE_SZ];
  const int lane   = threadIdx.x & 31;
  const int wid    = threadIdx.x >> 5;
  const int sample = blockIdx.x * WAVES_PER_BLOCK + wid;
  if (sample >= batch) return; // wave-uniform exit: active waves keep EXEC all-1s
  float* S = lds + wid * STATE_SZ;

  // |0...0> : lane owns slots {r*32 + lane}
#pragma unroll
  for (int r = 0; r < 32; ++r)
    S[r * 32 + lane] = (r == 0 && lane == 0) ? 1.0f : 0.0f;

  // Input-layer angles: theta_w = tanh(x[s,w]) (per-sample)
  float ci[N_QUBITS], si[N_QUBITS];
#pragma unroll
  for (int w = 0; w < N_QUBITS; ++w) {
    const float a = 0.5f * tanhf(x[sample * N_QUBITS + w]);
    ci[w] = cosf(a);
    si[w] = sinf(a);
  }
  apply_group4<0>(S, ci + 0, si + 0, lane); // wires 0-3  (bits 9..6)
  apply_group4<1>(S, ci + 4, si + 4, lane); // wires 4-7  (bits 5..2)
  apply_group2(S, ci[8], si[8], ci[9], si[9], lane); // wires 8-9 (bits 1..0)

  for (int l = 0; l < N_LAYERS; ++l) {
    float cl[N_QUBITS], sl[N_QUBITS];
#pragma unroll
    for (int w = 0; w < N_QUBITS; ++w) {
      const float a = 0.5f * params[l * N_QUBITS + w];
      cl[w] = cosf(a);
      sl[w] = sinf(a);
    }
    apply_group4<0>(S, cl + 0, sl + 0, lane);
    apply_group4<1>(S, cl + 4, sl + 4, lane);
    apply_group2(S, cl[8], sl[8], cl[9], sl[9], lane);

    // CNOT ring: gather through registers (same-wave LDS ops are in-order;
    // all 32 loads precede all 32 stores in program order)
    float tmp[32];
#pragma unroll
    for (int r = 0; r < 32; ++r)
      tmp[r] = S[cnot_src(r * 32 + lane)];
    asm volatile("" ::: "memory");
#pragma unroll
    for (int r = 0; r < 32; ++r)
      S[r * 32 + lane] = tmp[r];
  }

  // <Z_w> = sum_i S[i]^2 * (bit_{9-w}(i) ? -1 : +1)
  float acc[N_QUBITS];
#pragma unroll
  for (int w = 0; w < N_QUBITS; ++w) acc[w] = 0.0f;
#pragma unroll
  for (int r = 0; r < 32; ++r) {
    const int i = r * 32 + lane;
    const float v = S[i];
    const float p = v * v;
#pragma unroll
    for (int w = 0; w < N_QUBITS; ++w)
      acc[w] += ((i >> (9 - w)) & 1) ? -p : p;
  }
#pragma unroll
  for (int off = 16; off > 0; off >>= 1) {
#pragma unroll
    for (int w = 0; w < N_QUBITS; ++w)
      acc[w] += __shfl_xor(acc[w], off, 32);
  }
  if (lane == 0) {
#pragma unroll
    for (int w = 0; w < N_QUBITS; ++w)
      out[sample * N_QUBITS + w] = acc[w];
  }
}

extern "C" void kernel_launch(void* const* d_in, const int* in_sizes, int n_in,
                              void* d_out, int out_size, void* d_ws, size_t ws_size,
                              hipStream_t stream) {
  const float* x      = (const float*)d_in[0]; // [BATCH, 10] fp32
  const float* params = (const float*)d_in[1]; // [3, 10] fp32
  float* out          = (float*)d_out;         // [BATCH, 10] fp32
  const int batch = in_sizes[0] / N_QUBITS;
  const int blocks = (batch + WAVES_PER_BLOCK - 1) / WAVES_PER_BLOCK;
  qlayer_wmma_kernel<<<blocks, WAVES_PER_BLOCK * 32, 0, stream>>>(x, params, out, batch);
}